// TopKGate_9706626089980
// MI455X (gfx1250) — compile-verified
//
#include <hip/hip_runtime.h>
#include <hip/hip_bf16.h>
#include <math.h>

#define S_TOK 16384
#define MDIM  1024
#define NEXP  64
#define CAP   256

// output layout (floats, concatenated in reference return order)
#define OFF_LAUX  ((size_t)0)
#define CW_ELEMS  ((size_t)S_TOK * NEXP * CAP)        // 268435456
#define OFF_CW    ((size_t)1)
#define OFF_DM    (OFF_CW + CW_ELEMS)
#define OFF_EC    (OFF_DM + CW_ELEMS)
#define OFF_ALPHA (OFF_EC + NEXP)
#define OUT_TOTAL (OFF_ALPHA + S_TOK)                 // 536887361

typedef __attribute__((ext_vector_type(2))) float v2f;
typedef __attribute__((ext_vector_type(4))) float v4f;
typedef __attribute__((ext_vector_type(8))) float v8f;

// ---------------------------------------------------------------------------
// 0) zero-fill the 2.15 GB output with non-temporal 128-bit stores
// ---------------------------------------------------------------------------
__global__ void fill_zero_kernel(float* __restrict__ out) {
    size_t n4 = OUT_TOTAL / 4;
    size_t i = (size_t)blockIdx.x * blockDim.x + threadIdx.x;
    size_t stride = (size_t)gridDim.x * blockDim.x;
    v4f z = {0.f, 0.f, 0.f, 0.f};
    v4f* o4 = (v4f*)out;
    for (size_t j = i; j < n4; j += stride)
        __builtin_nontemporal_store(z, o4 + j);
    // tail elements
    for (size_t j = n4 * 4 + i; j < OUT_TOTAL; j += stride)
        out[j] = 0.f;
}

__global__ void ws_init_kernel(float* me, int* cnt) {
    int i = threadIdx.x;
    if (i < NEXP) { me[i] = 0.f; cnt[i] = 0; }
}

// ---------------------------------------------------------------------------
// 1) logits = x[S,M] * W[E,M]^T  via V_WMMA_F32_16X16X4_F32 (fp32, exact)
//    block = 128 threads = 4 waves; each wave owns one 16-row strip and
//    computes all 64 expert columns with 4 f32 accumulators.
// ---------------------------------------------------------------------------
__global__ void gate_gemm_kernel(const float* __restrict__ x,
                                 const float* __restrict__ w,
                                 float* __restrict__ logits) {
    const int wave = threadIdx.x >> 5;
    const int lane = threadIdx.x & 31;
    const int strip = blockIdx.x * 4 + wave;        // 0..1023
    const int row0 = strip * 16;
    const int rsel = lane & 15;                     // row (A) / col (B,D)
    const int kb   = (lane >> 4) << 1;              // K sub-offset {0,2}

    const float* ap  = x + (size_t)(row0 + rsel) * MDIM + kb;
    const float* bp0 = w + (size_t)rsel * MDIM + kb;          // cols 0..15
    const float* bp1 = bp0 + (size_t)16 * MDIM;               // cols 16..31
    const float* bp2 = bp0 + (size_t)32 * MDIM;               // cols 32..47
    const float* bp3 = bp0 + (size_t)48 * MDIM;               // cols 48..63

    v8f acc0 = {}; v8f acc1 = {}; v8f acc2 = {}; v8f acc3 = {};

    for (int k = 0; k < MDIM; k += 4) {
        if ((k & 255) == 0)                          // global_prefetch_b8 ahead
            __builtin_prefetch(ap + k + 512, 0, 0);
        v2f a  = *(const v2f*)(ap  + k);
        v2f b0 = *(const v2f*)(bp0 + k);
        v2f b1 = *(const v2f*)(bp1 + k);
        v2f b2 = *(const v2f*)(bp2 + k);
        v2f b3 = *(const v2f*)(bp3 + k);
        acc0 = __builtin_amdgcn_wmma_f32_16x16x4_f32(false, a, false, b0, (short)0, acc0, false, false);
        acc1 = __builtin_amdgcn_wmma_f32_16x16x4_f32(false, a, false, b1, (short)0, acc1, false, false);
        acc2 = __builtin_amdgcn_wmma_f32_16x16x4_f32(false, a, false, b2, (short)0, acc2, false, false);
        acc3 = __builtin_amdgcn_wmma_f32_16x16x4_f32(false, a, false, b3, (short)0, acc3, false, false);
    }

    // D layout: VGPR r -> row r (lanes 0-15) / row r+8 (lanes 16-31); N = lane&15
    const int mrow = row0 + ((lane >> 4) << 3);
    float* outp = logits + (size_t)mrow * NEXP + rsel;
#pragma unroll
    for (int r = 0; r < 8; ++r) {
        outp[(size_t)r * NEXP +  0] = acc0[r];
        outp[(size_t)r * NEXP + 16] = acc1[r];
        outp[(size_t)r * NEXP + 32] = acc2[r];
        outp[(size_t)r * NEXP + 48] = acc3[r];
    }
}

// ---------------------------------------------------------------------------
// 2) per-token softmax / argmax / alpha; accumulate me[e] and counts[e];
//    generate the RTS random value per token (hash-based uniform).
// ---------------------------------------------------------------------------
__global__ void gate_softmax_kernel(const float* __restrict__ logits,
                                    int* __restrict__ idx,
                                    float* __restrict__ randv,
                                    float* __restrict__ gatev,
                                    float* __restrict__ me,
                                    int* __restrict__ cnt,
                                    float* __restrict__ out_alpha) {
    __shared__ float sme[NEXP];
    __shared__ int   scnt[NEXP];
    const int tid = threadIdx.x;
    if (tid < NEXP) { sme[tid] = 0.f; scnt[tid] = 0; }
    __syncthreads();

    const int t = blockIdx.x * 256 + tid;
    const float* lp = logits + (size_t)t * NEXP;

    float lmax = -3.4e38f; int amax = 0;
#pragma unroll 8
    for (int e = 0; e < NEXP; ++e) {
        float v = lp[e];
        if (v > lmax) { lmax = v; amax = e; }
    }
    float sum = 0.f;
#pragma unroll 8
    for (int e = 0; e < NEXP; ++e) sum += __expf(lp[e] - lmax);
    const float inv = 1.0f / sum;

#pragma unroll 8
    for (int e = 0; e < NEXP; ++e)
        atomicAdd(&sme[e], __expf(lp[e] - lmax) * inv);
    atomicAdd(&scnt[amax], 1);

    const float alpha = inv;                 // exp(0)/sum == max gate
    idx[t] = amax;
    gatev[t] = alpha;
    out_alpha[t] = alpha;

    // hash-based uniform in [0,1) for random tie-break (use_rts)
    unsigned int h = (unsigned int)t;
    h ^= h >> 16; h *= 0x7feb352dU;
    h ^= h >> 15; h *= 0x846ca68bU;
    h ^= h >> 16;
    randv[t] = (float)(h >> 8) * (1.0f / 16777216.0f);

    __syncthreads();
    if (tid < NEXP) {
        atomicAdd(me + tid, sme[tid]);
        atomicAdd(cnt + tid, scnt[tid]);
    }
}

// ---------------------------------------------------------------------------
// 3) per-expert capacity threshold: binary search on rand so that
//    #(assigned && rand >= thr) <= CAP. thr = -1 means keep all.
// ---------------------------------------------------------------------------
__global__ void thresh_kernel(const int* __restrict__ idx,
                              const float* __restrict__ randv,
                              const int* __restrict__ cnt,
                              float* __restrict__ thr) {
    const int e = blockIdx.x;
    __shared__ int red[256];
    const int n = cnt[e];
    if (n <= CAP) { if (threadIdx.x == 0) thr[e] = -1.f; return; }

    float lo = 0.f, hi = 1.0f;
    for (int it = 0; it < 24; ++it) {
        float mid = 0.5f * (lo + hi);
        int c = 0;
        for (int t = threadIdx.x; t < S_TOK; t += 256)
            c += (idx[t] == e && randv[t] >= mid) ? 1 : 0;
        red[threadIdx.x] = c;
        __syncthreads();
        for (int off = 128; off; off >>= 1) {
            if (threadIdx.x < off) red[threadIdx.x] += red[threadIdx.x + off];
            __syncthreads();
        }
        int total = red[0];
        __syncthreads();
        if (total > CAP) lo = mid; else hi = mid;
    }
    if (threadIdx.x == 0) thr[e] = hi;
}

// ---------------------------------------------------------------------------
// 4) slot assignment: per-expert cumsum over token order of kept tokens.
//    One block per expert; 256-wide LDS inclusive scan per chunk.
// ---------------------------------------------------------------------------
__global__ void slots_kernel(const int* __restrict__ idx,
                             const float* __restrict__ randv,
                             const float* __restrict__ thr,
                             int* __restrict__ slot) {
    const int e = blockIdx.x;
    __shared__ int sc[256];
    __shared__ int running;
    const float t = thr[e];
    if (threadIdx.x == 0) running = 0;
    __syncthreads();

    for (int base = 0; base < S_TOK; base += 256) {
        const int tok = base + threadIdx.x;
        const bool mine = (idx[tok] == e);
        const int kept = (mine && (t < 0.f || randv[tok] >= t)) ? 1 : 0;
        sc[threadIdx.x] = kept;
        __syncthreads();
        for (int off = 1; off < 256; off <<= 1) {
            int v = (threadIdx.x >= off) ? sc[threadIdx.x - off] : 0;
            __syncthreads();
            sc[threadIdx.x] += v;
            __syncthreads();
        }
        const int pos = running + sc[threadIdx.x] - kept;   // exclusive prefix
        if (mine) slot[tok] = (kept && pos < CAP) ? pos : -1;
        __syncthreads();
        if (threadIdx.x == 0) running += sc[255];
        __syncthreads();
    }
}

// ---------------------------------------------------------------------------
// 5) scatter the <=16384 nonzeros into combine_weights / dispatch_mask
// ---------------------------------------------------------------------------
__global__ void scatter_kernel(const int* __restrict__ idx,
                               const int* __restrict__ slot,
                               const float* __restrict__ gatev,
                               float* __restrict__ cw,
                               float* __restrict__ dm) {
    const int t = blockIdx.x * 256 + threadIdx.x;
    const int s = slot[t];
    if (s >= 0) {
        const int e = idx[t];
        const size_t off = ((size_t)t * NEXP + e) * CAP + s;
        const float g = gatev[t];
        cw[off] = g;
        dm[off] = (g != 0.f) ? 1.0f : 0.0f;
    }
}

// ---------------------------------------------------------------------------
// 6) l_aux = E * sum_e (mean_gates[e] * frac_assigned[e]); exp_counts out
// ---------------------------------------------------------------------------
__global__ void finalize_kernel(const float* __restrict__ me,
                                const int* __restrict__ cnt,
                                float* __restrict__ out) {
    __shared__ float part[NEXP];
    const int e = threadIdx.x;   // 64 threads
    const float m = me[e] * (1.0f / (float)S_TOK);
    const float c = (float)cnt[e] * (1.0f / (float)S_TOK);
    part[e] = m * c * (float)NEXP;
    out[OFF_EC + e] = (float)cnt[e];
    __syncthreads();
    for (int off = 32; off; off >>= 1) {
        if (e < off) part[e] += part[e + off];
        __syncthreads();
    }
    if (e == 0) out[OFF_LAUX] = part[0];
}

// ---------------------------------------------------------------------------
extern "C" void kernel_launch(void* const* d_in, const int* in_sizes, int n_in,
                              void* d_out, int out_size, void* d_ws, size_t ws_size,
                              hipStream_t stream) {
    (void)in_sizes; (void)n_in; (void)out_size; (void)ws_size;

    const float* x  = (const float*)d_in[0];   // [S, M]
    const float* wg = (const float*)d_in[1];   // [E, M]
    float* out = (float*)d_out;

    // workspace layout
    char* wsb = (char*)d_ws;
    float* logits = (float*)(wsb);                                   // 4 MB
    int*   idx    = (int*)  (wsb + (size_t)4 * 1024 * 1024);         // 64 KB
    float* randv  = (float*)(wsb + (size_t)4 * 1024 * 1024 + 1 * 65536);
    float* gatev  = (float*)(wsb + (size_t)4 * 1024 * 1024 + 2 * 65536);
    int*   slot   = (int*)  (wsb + (size_t)4 * 1024 * 1024 + 3 * 65536);
    float* me     = (float*)(wsb + (size_t)4 * 1024 * 1024 + 4 * 65536);
    int*   cnt    = (int*)  (wsb + (size_t)4 * 1024 * 1024 + 4 * 65536 + 256);
    float* thr    = (float*)(wsb + (size_t)4 * 1024 * 1024 + 4 * 65536 + 512);

    // 0) zero the 2.15 GB output (dominant cost: ~92 us at 23.3 TB/s)
    fill_zero_kernel<<<8192, 256, 0, stream>>>(out);
    ws_init_kernel<<<1, 64, 0, stream>>>(me, cnt);

    // 1) fp32 WMMA gate GEMM: 1024 strips of 16 rows, 4 waves per block
    gate_gemm_kernel<<<S_TOK / 64, 128, 0, stream>>>(x, wg, logits);

    // 2) softmax / argmax / alpha / me / counts / rand
    gate_softmax_kernel<<<S_TOK / 256, 256, 0, stream>>>(logits, idx, randv, gatev,
                                                         me, cnt, out + OFF_ALPHA);

    // 3) per-expert capacity thresholds
    thresh_kernel<<<NEXP, 256, 0, stream>>>(idx, randv, cnt, thr);

    // 4) slot assignment (cumsum within each expert's kept set)
    slots_kernel<<<NEXP, 256, 0, stream>>>(idx, randv, thr, slot);

    // 5) scatter nonzeros into combine_weights / dispatch_mask
    scatter_kernel<<<S_TOK / 256, 256, 0, stream>>>(idx, slot, gatev,
                                                    out + OFF_CW, out + OFF_DM);

    // 6) l_aux and exp_counts
    finalize_kernel<<<1, 64, 0, stream>>>(me, cnt, out);
}